// CrossSetNorm_8581344657856
// MI455X (gfx1250) — compile-verified
//
#include <hip/hip_runtime.h>
#include <stdint.h>
#include <stddef.h>

// CrossSetNorm for MI455X (gfx1250).
// Memory-bound: 687MB in + 687MB out => ~59us floor @ 23.3 TB/s.
//
// Strategy (v2):
//  - one workgroup per (batch, segment); the segment tile (<=200KB) is staged
//    into LDS via CDNA5 async global->LDS DMA (ASYNCcnt-tracked).
//  - while the DMA is in flight, the stats pass reads x through the normal
//    vector path: both streams share cachelines, so L2 (192MB) absorbs the
//    duplicate read and HBM still sees x exactly once. This removes the
//    HBM-idle window that a fill->stats->store serial pipeline would have
//    at 1 WG/WGP occupancy.
//  - normalize pass runs out of the LDS tile (guaranteed on-WGP reuse) with
//    non-temporal stores (output is never re-read; keep L2 for the input).

static constexpr int kB     = 2048;
static constexpr int kStot  = 328;
static constexpr int kSobj  = 128;   // RO_MAX + 1
static constexpr int kSroad = 200;   // 328 - 128
static constexpr int kD     = 256;
#define EPS_ 1e-6f

template<int S, int OFF>
__device__ __forceinline__ void process_segment(const float* __restrict__ x,
                                                const int*   __restrict__ mask,
                                                const float* __restrict__ w,
                                                const float* __restrict__ bvec,
                                                float*       __restrict__ out,
                                                int bI, char* smem)
{
    float* aliveLds = (float*)smem;            // [<=256] floats (1KB region)
    float* tile     = (float*)(smem + 1024);   // [S][256] floats

    const int    tid     = threadIdx.x;
    const size_t rowBase = (size_t)bI * kStot + OFF;
    const float* gsrc    = x + rowBase * kD;

    // ---- Kick off async global->LDS DMA for the whole tile (ASYNCcnt) ----
    // Tile = S*64 16B chunks; 256 threads -> S/4 issues per thread
    // (road: 50/wave, obj: 32/wave; both under the 63-outstanding cap).
    {
        constexpr int nIter = S / 4;
        const char* gb = (const char*)gsrc;
#pragma unroll 4
        for (int i = 0; i < nIter; ++i) {
            const int      chunk   = tid + (i << 8);
            const uint32_t lds_off = 1024u + ((uint32_t)chunk << 4);
            const uint64_t ga      = (uint64_t)(uintptr_t)(gb + ((size_t)chunk << 4));
            asm volatile("global_load_async_to_lds_b128 %0, %1, off"
                         :: "v"(lds_off), "v"(ga) : "memory");
        }
    }

    // Alive flags (1.0 = alive, 0.0 = dead) while the DMA streams.
    if (tid < S) {
        aliveLds[tid] = (mask[rowBase + tid] == 0) ? 1.0f : 0.0f;
    }
    __syncthreads();   // alive flags visible; DMA still in flight (ASYNCcnt untouched)

    // ---- Pass 1 (overlapped with DMA): per-channel sum/sumsq + alive count.
    // Reads go through the vector path and hit the same lines the DMA is
    // pulling through L2, so HBM reads x only once.
    const int d = tid;
    float sum = 0.0f, sq = 0.0f, cnt = 0.0f;
#pragma unroll 8
    for (int s = 0; s < S; ++s) {
        const float a = aliveLds[s];
        const float v = gsrc[s * kD + d] * a;   // xm = x * alive
        sum += v;
        sq   = fmaf(v, v, sq);
        cnt += a;
    }

    // Reference edge-case semantics:
    //   counts = clip(cnt, 1); batch_ok = counts > 1
    //   mean   = sum/counts  (equals the reference's `where` in all cases)
    //   var sums over ALL S rows (dead rows contribute mean^2):
    //     sum_s (xm - mean)^2 = sq - 2*mean*sum + S*mean^2
    const float counts  = fmaxf(cnt, 1.0f);
    const bool  ok      = counts > 1.0f;
    const float mean    = sum / counts;
    float       var     = (sq - 2.0f * mean * sum + (float)S * mean * mean) / counts;
    var                 = ok ? var : 0.0f;
    const float inv_std = ok ? rsqrtf(var + EPS_) : 1.0f;

    const float scale = inv_std * w[d];
    const float shift = fmaf(-mean, scale, bvec[d]);   // (v - mean)*scale + b

    // ---- Drain the DMA, then normalize out of LDS with streaming stores ----
    asm volatile("s_wait_asynccnt 0x0" ::: "memory");
    __syncthreads();

    float* gout = out + rowBase * kD + d;
#pragma unroll 4
    for (int s = 0; s < S; ++s) {
        const float v = tile[s * kD + d] * aliveLds[s];
        __builtin_nontemporal_store(fmaf(v, scale, shift), gout + (size_t)s * kD);
    }
}

extern "C" __global__ __launch_bounds__(256)
void cross_set_norm_kernel(const float* __restrict__ x,
                           const int*   __restrict__ mask,   // True(!=0) = dead
                           const float* __restrict__ w_obj,
                           const float* __restrict__ b_obj,
                           const float* __restrict__ w_road,
                           const float* __restrict__ b_road,
                           float* __restrict__ out)
{
    extern __shared__ char smem[];
    const int wg  = blockIdx.x;
    const int bI  = wg >> 1;
    const int seg = wg & 1;
    if (seg == 0) {
        process_segment<kSobj, 0>(x, mask, w_obj, b_obj, out, bI, smem);
    } else {
        process_segment<kSroad, kSobj>(x, mask, w_road, b_road, out, bI, smem);
    }
}

extern "C" void kernel_launch(void* const* d_in, const int* in_sizes, int n_in,
                              void* d_out, int out_size, void* d_ws, size_t ws_size,
                              hipStream_t stream) {
    (void)in_sizes; (void)n_in; (void)out_size; (void)d_ws; (void)ws_size;
    const float* x    = (const float*)d_in[0];
    const int*   mask = (const int*)d_in[1];   // bool mask, integer on device
    const float* wo   = (const float*)d_in[2];
    const float* bo   = (const float*)d_in[3];
    const float* wr   = (const float*)d_in[4];
    const float* br   = (const float*)d_in[5];
    float*       out  = (float*)d_out;

    dim3 grid(kB * 2);     // 4096 workgroups: (batch, segment)
    dim3 block(256);       // 8 wave32s
    // 1KB alive region + road tile (200 * 256 * 4B) = 205,824 B dynamic LDS
    // (CDNA5 allows up to 320KB per workgroup).
    const size_t lds_bytes = 1024 + (size_t)kSroad * kD * sizeof(float);
    hipLaunchKernelGGL(cross_set_norm_kernel, grid, block, lds_bytes, stream,
                       x, mask, wo, bo, wr, br, out);
}